// EdgeNet_24936580121170
// MI455X (gfx1250) — compile-verified
//
#include <hip/hip_runtime.h>
#include <hip/hip_bf16.h>
#include <math.h>

#define N_NODES 8192
#define N_EDGES 32768
#define NCHUNKS 4
#define CHUNK   (N_NODES / NCHUNKS)   // 2048 nodes per chunk

typedef __attribute__((ext_vector_type(2))) float v2f;
typedef __attribute__((ext_vector_type(8))) float v8f;

// Branchless fast activations (no EXEC-divergent libm paths between WMMAs).
// tanh(x) = 1 - 2/(exp(2x)+1): exp overflow -> +1, underflow -> -1. Robust.
__device__ __forceinline__ float fast_tanh(float x) {
    const float e = __expf(2.0f * x);
    return 1.0f - 2.0f / (e + 1.0f);
}
__device__ __forceinline__ float fast_sigmoid(float x) {
    return 1.0f / (1.0f + __expf(-x));
}

// ---------------------------------------------------------------------------
// Kernel 1: stream Ri/Ro (2 GB total, the bandwidth roofline) and produce
// partial B = [Ro^T X | Ri^T X] per node-chunk into d_ws.
// Bpart layout: [chunk][edge][8] f32.
// Each thread owns 2 consecutive edges (float2 loads -> 256B per wave per
// load, fully coalesced). X chunk lives in LDS (32 KB of the 320 KB WGP LDS).
// ---------------------------------------------------------------------------
__global__ __launch_bounds__(256) void edgenet_incidence(
    const float* __restrict__ X, const float* __restrict__ Ri,
    const float* __restrict__ Ro, float* __restrict__ Bpart)
{
    __shared__ float4 Xs[CHUNK];
    const int tid   = threadIdx.x;
    const int nbase = blockIdx.y * CHUNK;
    const float4* X4 = (const float4*)X;
    for (int i = tid; i < CHUNK; i += 256) Xs[i] = X4[nbase + i];
    __syncthreads();

    const int e0 = blockIdx.x * 512 + tid * 2;
    float4 aco0 = {0.f,0.f,0.f,0.f}, aco1 = {0.f,0.f,0.f,0.f};
    float4 aci0 = {0.f,0.f,0.f,0.f}, aci1 = {0.f,0.f,0.f,0.f};

    #pragma unroll 4
    for (int n = 0; n < CHUNK; ++n) {
        const size_t rowoff = (size_t)(nbase + n) * N_EDGES + e0;
        const float2 ro = *(const float2*)(Ro + rowoff);
        const float2 ri = *(const float2*)(Ri + rowoff);
        const float4 x  = Xs[n];
        aco0.x += ro.x * x.x; aco0.y += ro.x * x.y;
        aco0.z += ro.x * x.z; aco0.w += ro.x * x.w;
        aco1.x += ro.y * x.x; aco1.y += ro.y * x.y;
        aco1.z += ro.y * x.z; aco1.w += ro.y * x.w;
        aci0.x += ri.x * x.x; aci0.y += ri.x * x.y;
        aci0.z += ri.x * x.z; aci0.w += ri.x * x.w;
        aci1.x += ri.y * x.x; aci1.y += ri.y * x.y;
        aci1.z += ri.y * x.z; aci1.w += ri.y * x.w;
    }

    // edge e0: cols 0..3 = Ro^T X, cols 4..7 = Ri^T X ; then edge e0+1
    float4* bp = (float4*)(Bpart + ((size_t)blockIdx.y * N_EDGES + e0) * 8);
    bp[0] = aco0; bp[1] = aci0;   // edge e0
    bp[2] = aco1; bp[3] = aci1;   // edge e0+1
}

// ---------------------------------------------------------------------------
// Kernel 2: MLP via V_WMMA_F32_16X16X4_F32.
// One wave32 handles a tile of 16 edges. K=8 -> two K=4 WMMA steps;
// N=100 -> 7 column tiles of 16 (zero-padded past col 99).
// Layer 2 (100 -> 1) done as register dot-product + shfl_xor reduction.
// ---------------------------------------------------------------------------
__global__ __launch_bounds__(256) void edgenet_mlp(
    const float* __restrict__ Bpart, const float* __restrict__ W1,
    const float* __restrict__ b1,    const float* __restrict__ W2,
    const float* __restrict__ b2,    float* __restrict__ out)
{
    const int lane  = threadIdx.x & 31;
    const int wave  = threadIdx.x >> 5;
    const int ebase = (blockIdx.x * 8 + wave) * 16;
    const int row   = lane & 15;      // A: M index / B: N index
    const int h     = lane >> 4;      // lane half selects K pair

    // Build A fragments (ISA layout: v0 = A[M, K=h*2], v1 = A[M, K=h*2+1]),
    // summing the node-chunk partials on the fly.
    v2f a0 = {0.f, 0.f}, a1 = {0.f, 0.f};
    const size_t eoff = (size_t)(ebase + row) * 8;
    #pragma unroll
    for (int c = 0; c < NCHUNKS; ++c) {
        const float* brow = Bpart + (size_t)c * N_EDGES * 8 + eoff;
        a0.x += brow[h * 2 + 0];
        a0.y += brow[h * 2 + 1];
        a1.x += brow[4 + h * 2 + 0];
        a1.y += brow[4 + h * 2 + 1];
    }

    float partial[8];
    #pragma unroll
    for (int r = 0; r < 8; ++r) partial[r] = 0.f;

    #pragma unroll
    for (int t = 0; t < 7; ++t) {
        const int   col = t * 16 + row;
        const float m   = (col < 100) ? 1.0f : 0.0f;   // branchless zero-pad
        const int   cc  = (col < 100) ? col  : 99;
        const int   k0  = h * 2;

        v2f bf0, bf1;                                  // B frag: [K=k, N=col]
        bf0.x = W1[(k0 + 0) * 100 + cc] * m;
        bf0.y = W1[(k0 + 1) * 100 + cc] * m;
        bf1.x = W1[(4 + k0 + 0) * 100 + cc] * m;
        bf1.y = W1[(4 + k0 + 1) * 100 + cc] * m;

        v8f c8 = {0.f,0.f,0.f,0.f,0.f,0.f,0.f,0.f};
        c8 = __builtin_amdgcn_wmma_f32_16x16x4_f32(
                 false, a0, false, bf0, (short)0, c8, false, false);
        c8 = __builtin_amdgcn_wmma_f32_16x16x4_f32(
                 false, a1, false, bf1, (short)0, c8, false, false);

        const float bias = b1[cc] * m;
        const float w2   = W2[cc] * m;
        #pragma unroll
        for (int r = 0; r < 8; ++r) {
            // D reg r, this lane = H[ebase + r + h*8, col]
            const float hv = fast_tanh(c8[r] + bias);
            partial[r] += hv * w2;                     // layer-2 dot product
        }
    }

    // Reduce layer-2 partials over the 16 lanes of each half (N dimension).
    #pragma unroll
    for (int mask = 1; mask < 16; mask <<= 1) {
        #pragma unroll
        for (int r = 0; r < 8; ++r)
            partial[r] += __shfl_xor(partial[r], mask, 32);
    }

    if (row == 0) {
        const float bb = b2[0];
        #pragma unroll
        for (int r = 0; r < 8; ++r)
            out[ebase + h * 8 + r] = fast_sigmoid(partial[r] + bb);
    }
}

extern "C" void kernel_launch(void* const* d_in, const int* in_sizes, int n_in,
                              void* d_out, int out_size, void* d_ws, size_t ws_size,
                              hipStream_t stream) {
    const float* X  = (const float*)d_in[0];   // [8192, 4]
    const float* Ri = (const float*)d_in[1];   // [8192, 32768]
    const float* Ro = (const float*)d_in[2];   // [8192, 32768]
    const float* W1 = (const float*)d_in[3];   // [8, 100]
    const float* b1 = (const float*)d_in[4];   // [100]
    const float* W2 = (const float*)d_in[5];   // [100, 1]
    const float* b2 = (const float*)d_in[6];   // [1]
    float* out   = (float*)d_out;              // [32768]
    float* Bpart = (float*)d_ws;               // NCHUNKS * 32768 * 8 f32 = 4 MB

    // Kernel 1: 64 edge-tiles x 4 node-chunks = 256 blocks (2048 waves).
    dim3 g1(N_EDGES / 512, NCHUNKS);
    edgenet_incidence<<<g1, 256, 0, stream>>>(X, Ri, Ro, Bpart);

    // Kernel 2: 8 waves/block * 16 edges/wave = 128 edges per block.
    edgenet_mlp<<<N_EDGES / 128, 256, 0, stream>>>(Bpart, W1, b1, W2, b2, out);
}